// Corner2Depth_1726576857094
// MI455X (gfx1250) — compile-verified
//
#include <hip/hip_runtime.h>

// Corner2Depth for MI455X (gfx1250, wave32).
// denom GEMM (HWx3 . 3xN) done with V_WMMA_F32_16X16X4_F32, two tiles per wave
// (16 rays x 16 walls each), transposed back to ray-major through padded LDS.
// Epilogue: scale = (-d) * v_rcp_f32(denom), xz bounds test (EPS pre-folded
// into the stored bounds), argmin over walls.

typedef __attribute__((ext_vector_type(2))) float v2f;
typedef __attribute__((ext_vector_type(8))) float v8f;

#define C2D_EPS   0.01f
#define MAXW      16     // WMMA tile covers up to 16 walls (N=12 here)
#define TPB       256    // 8 wave32s
#define LDS_PITCH 17     // 16 + 1 pad: avoids 2-way LDS bank conflict

__global__ __launch_bounds__(TPB)
void corner2depth_wmma_kernel(const float* __restrict__ corners, // (B, Nmax, 3)
                              const float* __restrict__ grid,    // (H*W, 3)
                              const int*   __restrict__ nums,    // (B,)
                              float* __restrict__ depth_out,     // (B, H*W)
                              float* __restrict__ nrm_out,       // (B, H*W, 3)
                              int Nmax, long long HW)
{
    __shared__ float s_nx[MAXW], s_nz[MAXW], s_nd[MAXW];        // nd = -d
    __shared__ float s_minx[MAXW], s_maxx[MAXW];                // min-EPS, max+EPS
    __shared__ float s_minz[MAXW], s_maxz[MAXW];
    __shared__ float s_denom[TPB * LDS_PITCH];                  // [rayLocal][wall]

    const int tid = threadIdx.x;
    const int b   = blockIdx.y;
    int n = nums[b];
    if (n > MAXW) n = MAXW;

    // ---- per-block wall setup (threads 0..15) --------------------------------
    if (tid < MAXW) {
        float nx = 0.f, nz = 0.f, nd = 0.f;
        float mnx = 0.f, mxx = 0.f, mnz = 0.f, mxz = 0.f;
        if (tid < n) {
            const long long base = ((long long)b * Nmax + tid) * 3;
            const int j = (tid + 1 == n) ? 0 : tid + 1;
            const long long basej = ((long long)b * Nmax + j) * 3;
            const float x0 = corners[base + 0], z0 = corners[base + 2];
            const float x1 = corners[basej + 0], z1 = corners[basej + 2];
            const float dx = x1 - x0, dz = z1 - z0;
            nx = -dz;                       // cross(diff, +y) = (-dz, 0, dx)
            nz =  dx;
            nd = nx * x0 + nz * z0;         // -d  (d = -<n, c0>)
            mnx = fminf(x0, x1) - C2D_EPS;  mxx = fmaxf(x0, x1) + C2D_EPS;
            mnz = fminf(z0, z1) - C2D_EPS;  mxz = fmaxf(z0, z1) + C2D_EPS;
        }
        s_nx[tid] = nx; s_nz[tid] = nz; s_nd[tid] = nd;
        s_minx[tid] = mnx; s_maxx[tid] = mxx;
        s_minz[tid] = mnz; s_maxz[tid] = mxz;
    }
    __syncthreads();

    // ---- load this thread's ray (n_y == 0 so g_y is never needed) -----------
    const long long p = (long long)blockIdx.x * TPB + tid;  // ray index in batch
    const float gx = grid[p * 3 + 0];
    const float gz = grid[p * 3 + 2];

    // ---- WMMA: denom[ray][wall] = gx*nx + gz*nz ------------------------------
    // A (16x4, f32): lanes 0-15 hold M=lane, VGPR0/1 = K0/K1; lanes 16-31 = K2/K3.
    //   K rows are (gx, 0, gz, 0).
    // B (4x16, f32): column N = lane&15; VGPR0 = K0 (lanes<16) / K2 (lanes>=16).
    //   K rows are (nx, 0, nz, 0).
    // D (16x16, f32): lane L, vgpr r -> (M = r + 8*(L>>4), N = L&15).
    const int lane = tid & 31;
    const int lh   = lane & 15;
    const int hlf  = lane >> 4;
    const int wv   = tid >> 5;

    // rows of tile 0 are the wave's rays 0..15, tile 1 are rays 16..31
    const float gx0 = __shfl(gx, lh,      32);
    const float gz0 = __shfl(gz, lh,      32);
    const float gx1 = __shfl(gx, lh + 16, 32);
    const float gz1 = __shfl(gz, lh + 16, 32);

    v2f a0, a1, bm;
    a0.x = hlf ? gz0 : gx0;  a0.y = 0.f;
    a1.x = hlf ? gz1 : gx1;  a1.y = 0.f;
    bm.x = hlf ? s_nz[lh] : s_nx[lh];  bm.y = 0.f;

    v8f acc0 = {}, acc1 = {};
    acc0 = __builtin_amdgcn_wmma_f32_16x16x4_f32(false, a0, false, bm,
                                                 (short)0, acc0, false, false);
    acc1 = __builtin_amdgcn_wmma_f32_16x16x4_f32(false, a1, false, bm,
                                                 (short)0, acc1, false, false);

    // transpose D back to ray-major through LDS
    const int mBase = hlf * 8;
#pragma unroll
    for (int r = 0; r < 8; ++r) {
        s_denom[(wv * 32 +      mBase + r) * LDS_PITCH + lh] = acc0[r];
        s_denom[(wv * 32 + 16 + mBase + r) * LDS_PITCH + lh] = acc1[r];
    }
    __syncthreads();

    // ---- epilogue: scale, bounds test, argmin --------------------------------
    float best = __builtin_inff();
    int   bi   = 0;
    for (int i = 0; i < n; ++i) {
        const float denom = s_denom[tid * LDS_PITCH + i];
        const float scale = s_nd[i] * __builtin_amdgcn_rcpf(denom);
        const float ix = gx * scale;
        const float iz = gz * scale;
        const bool ok = (ix <= s_maxx[i]) & (ix >= s_minx[i])
                      & (iz <= s_maxz[i]) & (iz >= s_minz[i])
                      & (scale > 0.f);
        if (ok && scale < best) { best = scale; bi = i; }   // first-min, like argmin
    }

    const long long o = (long long)b * HW + p;
    depth_out[o]       = best;            // +inf if no wall hit (matches reference)
    nrm_out[o * 3 + 0] = s_nx[bi];
    nrm_out[o * 3 + 1] = 0.f;
    nrm_out[o * 3 + 2] = s_nz[bi];
}

extern "C" void kernel_launch(void* const* d_in, const int* in_sizes, int n_in,
                              void* d_out, int out_size, void* d_ws, size_t ws_size,
                              hipStream_t stream) {
    const float* corners = (const float*)d_in[0];   // (B, N, 3) f32
    const float* grid    = (const float*)d_in[1];   // (1, H, W, 3) f32
    const int*   nums    = (const int*)d_in[2];     // (B,) i32

    const int       B    = in_sizes[2];
    const int       Nmax = in_sizes[0] / (B * 3);
    const long long HW   = in_sizes[1] / 3;         // 512*1024, multiple of TPB

    float* depth = (float*)d_out;                   // (B,1,H,W) flat
    float* nrm   = depth + (long long)B * HW;       // (B,H,W,3) flat

    dim3 grd((unsigned)(HW / TPB), (unsigned)B);
    corner2depth_wmma_kernel<<<grd, TPB, 0, stream>>>(corners, grid, nums,
                                                      depth, nrm, Nmax, HW);
}